// LaneATT_81930796138612
// MI455X (gfx1250) — compile-verified
//
#include <hip/hip_runtime.h>

typedef __attribute__((ext_vector_type(2))) float v2f;
typedef __attribute__((ext_vector_type(8))) float v8f;

#define BATCH 32
#define CIN   512
#define HH    11
#define WW    20
#define NA    1000
#define CAC   64
#define FEATC 704          // CAC*HH
#define HEADN 75           // 2 cls + 73 reg
#define OUTC  77           // 2 + 2 + 73
#define BSC_LD 1008        // padded row stride for transposed att_w (16B-aligned rows)
#define BH_LD  80          // padded row stride for transposed head weights

// ---------------------------------------------------------------------------
// CDNA5 async global->LDS copy (ASYNCcnt path). The low 32 bits of a generic
// pointer into LDS are the wave-relative LDS byte offset (ISA aperture rule),
// which is exactly what the VDST operand expects.
// ---------------------------------------------------------------------------
__device__ __forceinline__ void async_copy_b128(void* lds_ptr, const void* gptr) {
  unsigned loff = (unsigned)(unsigned long long)lds_ptr;
  asm volatile("global_load_async_to_lds_b128 %0, %1, off"
               :
               : "v"(loff), "v"((unsigned long long)gptr)
               : "memory");
}
__device__ __forceinline__ void wait_async0() {
  asm volatile("s_wait_asynccnt 0x0" ::: "memory");
}

// ---------------------------------------------------------------------------
// 1x1 conv: feats[b][o][y][x] = sum_c x[b,c,y,x]*w[o,c] + bias[o]
// ---------------------------------------------------------------------------
__global__ __launch_bounds__(256)
void conv1x1_kernel(const float* __restrict__ x, const float* __restrict__ w,
                    const float* __restrict__ bias, float* __restrict__ feats) {
  int idx = blockIdx.x * blockDim.x + threadIdx.x;       // over BATCH*CAC*HH*WW
  const int total = BATCH * CAC * HH * WW;
  if (idx >= total) return;
  int hw = idx % (HH * WW);
  int o  = (idx / (HH * WW)) % CAC;
  int b  = idx / (CAC * HH * WW);
  const float* xp = x + (size_t)b * CIN * HH * WW + hw;
  const float* wp = w + (size_t)o * CIN;
  float acc = bias[o];
#pragma unroll 8
  for (int c = 0; c < CIN; ++c) acc += xp[(size_t)c * HH * WW] * wp[c];
  feats[idx] = acc;
}

// ---------------------------------------------------------------------------
// gather: feat[b*NA+n][f] = invalid ? 0 : feats[b][ca][y][cut_xs]
// also writes the raw-feature half of cat (columns FEATC..2*FEATC-1)
// ---------------------------------------------------------------------------
__global__ __launch_bounds__(256)
void gather_kernel(const float* __restrict__ feats, const int* __restrict__ cut_xs,
                   const int* __restrict__ invalid, float* __restrict__ feat,
                   float* __restrict__ cat) {
  size_t idx = (size_t)blockIdx.x * blockDim.x + threadIdx.x;  // over BATCH*NA*FEATC
  const size_t total = (size_t)BATCH * NA * FEATC;
  if (idx >= total) return;
  int f = (int)(idx % FEATC);
  int n = (int)((idx / FEATC) % NA);
  int b = (int)(idx / ((size_t)FEATC * NA));
  int ca = f / HH;
  int y  = f % HH;
  int ci = n * FEATC + f;                        // (n, ca, y) flat
  float v = 0.0f;
  if (!invalid[ci]) {
    int xs = cut_xs[ci];
    v = feats[((size_t)(b * CAC + ca) * HH + y) * WW + xs];
  }
  size_t row = (size_t)b * NA + n;
  feat[row * FEATC + f] = v;
  cat[row * (2 * FEATC) + FEATC + f] = v;
}

// ---------------------------------------------------------------------------
// weight prep: transpose att_w -> Bsc[K=704][BSC_LD]; stack+transpose
// cls_w/reg_w -> Bh[K=1408][BH_LD]; bias75
// ---------------------------------------------------------------------------
__global__ __launch_bounds__(256)
void prep_kernel(const float* __restrict__ att_w, const float* __restrict__ cls_w,
                 const float* __restrict__ reg_w, const float* __restrict__ cls_b,
                 const float* __restrict__ reg_b, float* __restrict__ Bsc,
                 float* __restrict__ Bh, float* __restrict__ b75) {
  int idx = blockIdx.x * blockDim.x + threadIdx.x;
  if (idx < (NA - 1) * FEATC) {
    int n = idx / FEATC, k = idx % FEATC;
    Bsc[(size_t)k * BSC_LD + n] = att_w[idx];
  }
  if (idx < HEADN * 2 * FEATC) {
    int n = idx / (2 * FEATC), k = idx % (2 * FEATC);
    Bh[(size_t)k * BH_LD + n] =
        (n < 2) ? cls_w[n * 2 * FEATC + k] : reg_w[(n - 2) * 2 * FEATC + k];
  }
  if (idx < HEADN) b75[idx] = (idx < 2) ? cls_b[idx] : reg_b[idx - 2];
}

// ---------------------------------------------------------------------------
// Generic tiled fp32 WMMA GEMM:  C = A[M,K] * B[K,N] (+bias[N])
// block tile 128(M) x 64(N), K-step 16; 8 waves, each wave 32x32 via
// 2x2 grid of v_wmma_f32_16x16x4_f32 fragments. grid.z = batch.
// Global->LDS staging uses GLOBAL_LOAD_ASYNC_TO_LDS_B128 for in-range
// float4s; edge threads zero-fill / scalar-fill LDS directly.
// ---------------------------------------------------------------------------
#define BM 128
#define BN 64
#define KT 16
#define AS_LD 20   // 128x20 floats, rows 80B (16B aligned), conflict-free frag reads
#define BS_LD 72   // 16x72 floats, rows 288B

__global__ __launch_bounds__(256)
void wmma_gemm_kernel(const float* __restrict__ A, long long lda, long long strideA,
                      const float* __restrict__ Bm, long long ldb, long long strideB,
                      float* __restrict__ C, long long ldc, long long strideC,
                      const float* __restrict__ bias, int M, int N, int K) {
  __shared__ __align__(16) float As[BM * AS_LD];
  __shared__ __align__(16) float Bs[KT * BS_LD];

  const int tid  = threadIdx.x;
  const int lane = tid & 31;
  const int wave = tid >> 5;
  const int bz   = blockIdx.z;
  A  += (size_t)bz * strideA;
  Bm += (size_t)bz * strideB;
  C  += (size_t)bz * strideC;

  const int tileM = blockIdx.y * BM;
  const int tileN = blockIdx.x * BN;
  const int wm    = (wave & 3) * 32;   // wave M offset inside block tile
  const int wn    = (wave >> 2) * 32;  // wave N offset inside block tile
  const int half  = lane >> 4;         // 0 | 1  (lane group)
  const int half2 = half * 2;
  const int l16   = lane & 15;

  v8f acc[2][2];
  const v8f vzero = {0, 0, 0, 0, 0, 0, 0, 0};
#pragma unroll
  for (int im = 0; im < 2; ++im)
#pragma unroll
    for (int in = 0; in < 2; ++in) acc[im][in] = vzero;

  for (int kb = 0; kb < K; kb += KT) {
    // ---- stage A tile: 128x16 = 512 float4, 2 per thread (async to LDS)
#pragma unroll
    for (int q = 0; q < 2; ++q) {
      int t   = tid + q * 256;
      int row = t >> 2;
      int kc  = (t & 3) * 4;
      int gr  = tileM + row;
      int gk  = kb + kc;
      float* lp = &As[row * AS_LD + kc];
      if (gr < M && gk < K) {
        async_copy_b128(lp, A + (size_t)gr * lda + gk);
      } else {
        *(float4*)lp = make_float4(0.f, 0.f, 0.f, 0.f);
      }
    }
    // ---- stage B tile: 16x64 = 256 float4, 1 per thread (async to LDS)
    {
      int krow = tid >> 4;
      int nc   = (tid & 15) * 4;
      int gk   = kb + krow;
      int gn   = tileN + nc;
      float* lp = &Bs[krow * BS_LD + nc];
      if (gk < K && gn + 3 < N) {
        async_copy_b128(lp, Bm + (size_t)gk * ldb + gn);
      } else {
        float4 v = make_float4(0.f, 0.f, 0.f, 0.f);
        if (gk < K) {
          const float* bp = Bm + (size_t)gk * ldb + gn;
          if (gn + 0 < N) v.x = bp[0];
          if (gn + 1 < N) v.y = bp[1];
          if (gn + 2 < N) v.z = bp[2];
        }
        *(float4*)lp = v;
      }
    }
    wait_async0();
    __syncthreads();

#pragma unroll
    for (int kk = 0; kk < KT; kk += 4) {
      // A fragments (16x4): lanes 0-15 hold K=kk,kk+1; lanes 16-31 K=kk+2,kk+3
      v2f afrag[2];
#pragma unroll
      for (int im = 0; im < 2; ++im) {
        int m = wm + im * 16 + l16;
        afrag[im] = *(const v2f*)(&As[m * AS_LD + kk + half2]);
      }
      // B fragments (4x16): VGPR r holds K = kk + r + half*2, N = l16
      v2f bfrag[2];
#pragma unroll
      for (int in = 0; in < 2; ++in) {
        int n = wn + in * 16 + l16;
        v2f bf;
        bf[0] = Bs[(kk + half2 + 0) * BS_LD + n];
        bf[1] = Bs[(kk + half2 + 1) * BS_LD + n];
        bfrag[in] = bf;
      }
#pragma unroll
      for (int im = 0; im < 2; ++im)
#pragma unroll
        for (int in = 0; in < 2; ++in)
          acc[im][in] = __builtin_amdgcn_wmma_f32_16x16x4_f32(
              false, afrag[im], false, bfrag[in], (short)0, acc[im][in],
              false, false);
    }
    __syncthreads();
  }

  // ---- epilogue: C/D layout -> VGPR r: row = r + half*8, col = l16
#pragma unroll
  for (int im = 0; im < 2; ++im) {
#pragma unroll
    for (int in = 0; in < 2; ++in) {
      int col = tileN + wn + in * 16 + l16;
      if (col >= N) continue;
      float bv = bias ? bias[col] : 0.0f;
#pragma unroll
      for (int r = 0; r < 8; ++r) {
        int row = tileM + wm + im * 16 + half * 8 + r;
        if (row < M) C[(size_t)row * ldc + col] = acc[im][in][r] + bv;
      }
    }
  }
}

// ---------------------------------------------------------------------------
// row softmax over 999 scores + off-diagonal scatter into AM[b][i][0..999]
// ---------------------------------------------------------------------------
__global__ __launch_bounds__(256)
void softmax_att_kernel(const float* __restrict__ scores, float* __restrict__ am) {
  int row = blockIdx.x;                // b*NA + i
  int i   = row % NA;
  const float* s = scores + (size_t)row * (NA - 1);
  __shared__ float red[256];
  float mx = -1e30f;
  for (int j = threadIdx.x; j < NA - 1; j += 256) mx = fmaxf(mx, s[j]);
  red[threadIdx.x] = mx;
  __syncthreads();
  for (int st = 128; st > 0; st >>= 1) {
    if (threadIdx.x < st) red[threadIdx.x] = fmaxf(red[threadIdx.x], red[threadIdx.x + st]);
    __syncthreads();
  }
  mx = red[0];
  __syncthreads();
  float sum = 0.0f;
  for (int j = threadIdx.x; j < NA - 1; j += 256) sum += __expf(s[j] - mx);
  red[threadIdx.x] = sum;
  __syncthreads();
  for (int st = 128; st > 0; st >>= 1) {
    if (threadIdx.x < st) red[threadIdx.x] += red[threadIdx.x + st];
    __syncthreads();
  }
  float inv = 1.0f / red[0];
  float* o = am + (size_t)row * NA;
  for (int j = threadIdx.x; j < NA; j += 256) {
    float v = 0.0f;
    if (j != i) {
      int src = j - (j > i);
      v = __expf(s[src] - mx) * inv;
    }
    o[j] = v;
  }
}

// ---------------------------------------------------------------------------
// assemble reg_proposals: [cls(2) | anchor[2:4] | anchor[4:] + reg(73)]
// ---------------------------------------------------------------------------
__global__ __launch_bounds__(256)
void assemble_kernel(const float* __restrict__ headout, const float* __restrict__ anchors,
                     float* __restrict__ out1) {
  size_t idx = (size_t)blockIdx.x * blockDim.x + threadIdx.x;  // over 32000*77
  const size_t total = (size_t)BATCH * NA * OUTC;
  if (idx >= total) return;
  int c = (int)(idx % OUTC);
  size_t rn = idx / OUTC;          // b*NA + n
  int n = (int)(rn % NA);
  float v;
  if (c < 2)       v = headout[rn * HEADN + c];
  else if (c < 4)  v = anchors[n * OUTC + c];
  else             v = anchors[n * OUTC + c] + headout[rn * HEADN + (c - 2)];
  out1[idx] = v;
}

// ---------------------------------------------------------------------------
extern "C" void kernel_launch(void* const* d_in, const int* in_sizes, int n_in,
                              void* d_out, int out_size, void* d_ws, size_t ws_size,
                              hipStream_t stream) {
  const float* x      = (const float*)d_in[0];
  const float* conv_w = (const float*)d_in[1];
  const float* conv_b = (const float*)d_in[2];
  const float* att_w  = (const float*)d_in[3];
  const float* att_b  = (const float*)d_in[4];
  const float* cls_w  = (const float*)d_in[5];
  const float* cls_b  = (const float*)d_in[6];
  const float* reg_w  = (const float*)d_in[7];
  const float* reg_b  = (const float*)d_in[8];
  const float* anchors= (const float*)d_in[9];
  const int*   cut_xs = (const int*)d_in[12];
  const int*   invalid= (const int*)d_in[13];

  // workspace layout (floats); all region bases are 16B-aligned
  float* ws      = (float*)d_ws;
  float* feats   = ws;                                              // 450,560
  float* feat    = feats   + (size_t)BATCH * CAC * HH * WW;         // 22,528,000
  float* cat     = feat    + (size_t)BATCH * NA * FEATC;            // 45,056,000
  float* scores  = cat     + (size_t)BATCH * NA * 2 * FEATC;        // 31,968,000
  float* Bsc     = scores  + (size_t)BATCH * NA * (NA - 1);         // 704*1008
  float* Bh      = Bsc     + (size_t)FEATC * BSC_LD;                // 1408*80
  float* b75     = Bh      + (size_t)2 * FEATC * BH_LD;             // 128 (padded)
  float* headout = b75 + 128;                                       // 2,400,000

  float* out1 = (float*)d_out;                                      // [32000,77]
  float* am   = out1 + (size_t)BATCH * NA * OUTC;                   // [B,N,N]

  // 1) 1x1 conv
  {
    int total = BATCH * CAC * HH * WW;
    conv1x1_kernel<<<(total + 255) / 256, 256, 0, stream>>>(x, conv_w, conv_b, feats);
  }
  // 2) gather -> feat + cat right half
  {
    size_t total = (size_t)BATCH * NA * FEATC;
    gather_kernel<<<(int)((total + 255) / 256), 256, 0, stream>>>(feats, cut_xs, invalid,
                                                                  feat, cat);
  }
  // 3) weight transposes
  {
    int total = (NA - 1) * FEATC;   // biggest of the three jobs
    prep_kernel<<<(total + 255) / 256, 256, 0, stream>>>(att_w, cls_w, reg_w, cls_b,
                                                         reg_b, Bsc, Bh, b75);
  }
  // 4) scores = feat @ att_w.T + att_b : M=32000, N=999, K=704
  {
    dim3 grid((NA - 1 + BN - 1) / BN, (BATCH * NA + BM - 1) / BM, 1);
    wmma_gemm_kernel<<<grid, 256, 0, stream>>>(feat, FEATC, 0, Bsc, BSC_LD, 0,
                                               scores, NA - 1, 0, att_b,
                                               BATCH * NA, NA - 1, FEATC);
  }
  // 5) softmax + attention-matrix scatter (written straight into d_out)
  softmax_att_kernel<<<BATCH * NA, 256, 0, stream>>>(scores, am);

  // 6) att_feat[b] = AM[b] @ feat[b] -> cat left half : per batch M=1000,N=704,K=1000
  {
    dim3 grid((FEATC + BN - 1) / BN, (NA + BM - 1) / BM, BATCH);
    wmma_gemm_kernel<<<grid, 256, 0, stream>>>(
        am, NA, (long long)NA * NA,                 // A: AM[b], lda=1000
        feat, FEATC, (long long)NA * FEATC,         // B: feat[b], ldb=704
        cat, 2 * FEATC, (long long)NA * 2 * FEATC,  // C: cat[b] cols 0..703
        nullptr, NA, FEATC, NA);
  }
  // 7) head: headout = cat @ [cls_w;reg_w].T + bias : M=32000, N=75, K=1408
  {
    dim3 grid((HEADN + BN - 1) / BN, (BATCH * NA + BM - 1) / BM, 1);
    wmma_gemm_kernel<<<grid, 256, 0, stream>>>(cat, 2 * FEATC, 0, Bh, BH_LD, 0,
                                               headout, HEADN, 0, b75,
                                               BATCH * NA, HEADN, 2 * FEATC);
  }
  // 8) assemble proposals
  {
    size_t total = (size_t)BATCH * NA * OUTC;
    assemble_kernel<<<(int)((total + 255) / 256), 256, 0, stream>>>(headout, anchors, out1);
  }
}